// LowRankRayTracer_38749194945147
// MI455X (gfx1250) — compile-verified
//
#include <hip/hip_runtime.h>

typedef __attribute__((ext_vector_type(2))) float v2f;
typedef __attribute__((ext_vector_type(8))) float v8f;

#define D_DIRS   4096
#define K_PTS    128
#define R_RANK   32
#define F_SUB    8192
#define N_ROWS   (D_DIRS * K_PTS)     /* 524288 rows of 32 complex */
#define NCHUNK   (N_ROWS / 4)         /* 131072 K-chunks of 4 */
#define NB1      256                  /* phase-1 blocks */
#define WPB      8                    /* waves per block (256 thr) */
#define TOTAL_WAVES (NB1 * WPB)       /* 2048 -> 64 chunks/wave */
#define PARTIAL_FLOATS 2048           /* 8 tiles * 256 floats per block */
#define MOFF     (NB1 * PARTIAL_FLOATS)

#define WMMA4(A, B, C) \
  __builtin_amdgcn_wmma_f32_16x16x4_f32(false, (A), false, (B), (short)0, (C), false, false)

// ---------------------------------------------------------------------------
// Phase 1: per-block partial of the 32x32 complex Gram matrix via f32 WMMA.
// acc[0..3] = Re tiles (TL,TR,BL,BR) of Rr^T*Ar - Ri^T*Ai
// acc[4..7] = tiles of Rr^T*Ai + Ri^T*Ar   (final Mim = -this)
// ---------------------------------------------------------------------------
__global__ __launch_bounds__(256) void lrrt_partial_kernel(
    const float2* __restrict__ rad2,   // (N_ROWS, 32) complex, A-side (index i)
    const float2* __restrict__ att2,   // (N_ROWS, 32) complex, B-side (index j)
    float* __restrict__ ws) {
  const int lane   = threadIdx.x & 31;
  const int wv     = threadIdx.x >> 5;
  const int gw     = blockIdx.x * WPB + wv;
  const int col    = lane & 15;
  const int rowoff = (lane >> 4) << 1;        // lanes 0-15 -> K=0/1, 16-31 -> K=2/3

  v8f acc[8];
  #pragma unroll
  for (int t = 0; t < 8; ++t) acc[t] = (v8f){0.f,0.f,0.f,0.f,0.f,0.f,0.f,0.f};

  for (int c = gw; c < NCHUNK; c += TOTAL_WAVES) {
    const long base = (long)(c * 4 + rowoff) * 32;   // complex-element index of this lane's K=0 row
    // radiation (A matrix): top tile col=i, bottom tile col=i+16; two K rows per lane
    float2 rT0 = rad2[base + col];
    float2 rT1 = rad2[base + 32 + col];
    float2 rB0 = rad2[base + col + 16];
    float2 rB1 = rad2[base + 32 + col + 16];
    // attenuation (B matrix): left tile j=col, right tile j=col+16 (mirrored layout)
    float2 aL0 = att2[base + col];
    float2 aL1 = att2[base + 32 + col];
    float2 aR0 = att2[base + col + 16];
    float2 aR1 = att2[base + 32 + col + 16];

    v2f RrT = {rT0.x, rT1.x}, RiT = {rT0.y, rT1.y};
    v2f RrB = {rB0.x, rB1.x}, RiB = {rB0.y, rB1.y};
    v2f nRiT = -RiT, nRiB = -RiB;               // f32 WMMA has no A-negate; do it in VALU
    v2f ArL = {aL0.x, aL1.x}, AiL = {aL0.y, aL1.y};
    v2f ArR = {aR0.x, aR1.x}, AiR = {aR0.y, aR1.y};

    // Re tiles: Rr*Ar - Ri*Ai
    acc[0] = WMMA4(RrT, ArL, acc[0]);  acc[0] = WMMA4(nRiT, AiL, acc[0]);
    acc[1] = WMMA4(RrT, ArR, acc[1]);  acc[1] = WMMA4(nRiT, AiR, acc[1]);
    acc[2] = WMMA4(RrB, ArL, acc[2]);  acc[2] = WMMA4(nRiB, AiL, acc[2]);
    acc[3] = WMMA4(RrB, ArR, acc[3]);  acc[3] = WMMA4(nRiB, AiR, acc[3]);
    // Im(P) tiles: Rr*Ai + Ri*Ar
    acc[4] = WMMA4(RrT, AiL, acc[4]);  acc[4] = WMMA4(RiT, ArL, acc[4]);
    acc[5] = WMMA4(RrT, AiR, acc[5]);  acc[5] = WMMA4(RiT, ArR, acc[5]);
    acc[6] = WMMA4(RrB, AiL, acc[6]);  acc[6] = WMMA4(RiB, ArL, acc[6]);
    acc[7] = WMMA4(RrB, AiR, acc[7]);  acc[7] = WMMA4(RiB, ArR, acc[7]);
  }

  // Deterministic cross-wave reduction: waves 1..7 stage in LDS, wave 0 sums.
  __shared__ float lds[(WPB - 1) * PARTIAL_FLOATS];   // 56 KB
  if (wv > 0) {
    float* p = &lds[(wv - 1) * PARTIAL_FLOATS];
    #pragma unroll
    for (int t = 0; t < 8; ++t)
      #pragma unroll
      for (int v = 0; v < 8; ++v)
        p[t * 256 + v * 32 + lane] = acc[t][v];
  }
  __syncthreads();
  if (wv == 0) {
    #pragma unroll
    for (int t = 0; t < 8; ++t)
      #pragma unroll
      for (int v = 0; v < 8; ++v) {
        float s = acc[t][v];
        for (int w = 0; w < WPB - 1; ++w)
          s += lds[w * PARTIAL_FLOATS + t * 256 + v * 32 + lane];
        ws[(size_t)blockIdx.x * PARTIAL_FLOATS + t * 256 + v * 32 + lane] = s;
      }
  }
}

// ---------------------------------------------------------------------------
// Phase 1b: sum the 256 block partials (deterministic order), remap the WMMA
// fragment layout (VGPR v, lane L -> M = v + 8*(L>=16), N = L%16) to a
// canonical 32x32 matrix, apply conjugation sign: Mim = -Im(P).
// ---------------------------------------------------------------------------
__global__ __launch_bounds__(256) void lrrt_reduce_kernel(float* __restrict__ ws) {
  for (int pos = threadIdx.x; pos < PARTIAL_FLOATS; pos += 256) {
    float s = 0.f;
    for (int b = 0; b < NB1; ++b)
      s += ws[(size_t)b * PARTIAL_FLOATS + pos];
    const int t = pos >> 8, r = pos & 255, v = r >> 5, L = r & 31;
    const int part = t >> 2, ih = (t >> 1) & 1, jh = t & 1;
    const int i = ih * 16 + v + ((L >> 4) << 3);
    const int j = jh * 16 + (L & 15);
    if (part == 0) ws[MOFF + i * 32 + j]        = s;    // Mre
    else           ws[MOFF + 1024 + i * 32 + j] = -s;   // Mim = -Im(P)
  }
}

// ---------------------------------------------------------------------------
// Phase 2: csi[f] = scale * v_f^T M v_f  (complex), M cached in LDS.
// ---------------------------------------------------------------------------
__global__ __launch_bounds__(256) void lrrt_csi_kernel(
    const float2* __restrict__ vf,    // (F, 32) complex
    const float* __restrict__ M,      // ws+MOFF: Mre[1024] then Mim[1024]
    float* __restrict__ out) {
  __shared__ float sM[2048];
  for (int p = threadIdx.x; p < 2048; p += 256) sM[p] = M[p];
  __syncthreads();

  const int f = blockIdx.x * 256 + threadIdx.x;
  float2 v[32];
  #pragma unroll
  for (int j = 0; j < 32; ++j) v[j] = vf[(size_t)f * 32 + j];

  float cr = 0.f, ci = 0.f;
  for (int i = 0; i < 32; ++i) {
    float tr = 0.f, ti = 0.f;
    #pragma unroll
    for (int j = 0; j < 32; ++j) {
      const float mr = sM[i * 32 + j];
      const float mi = sM[1024 + i * 32 + j];
      tr = fmaf(mr, v[j].x, fmaf(-mi, v[j].y, tr));
      ti = fmaf(mr, v[j].y, fmaf( mi, v[j].x, ti));
    }
    cr = fmaf(v[i].x, tr, fmaf(-v[i].y, ti, cr));
    ci = fmaf(v[i].x, ti, fmaf( v[i].y, tr, ci));
  }
  const float scale = 0.0003814697265625f;   // (200/128)/4096
  out[2 * f]     = cr * scale;
  out[2 * f + 1] = ci * scale;
}

extern "C" void kernel_launch(void* const* d_in, const int* in_sizes, int n_in,
                              void* d_out, int out_size, void* d_ws, size_t ws_size,
                              hipStream_t stream) {
  const float2* att = (const float2*)d_in[0];   // attenuation_vectors (D,K,R) c64
  const float2* rad = (const float2*)d_in[1];   // radiation_vectors   (D,K,R) c64
  const float2* vfb = (const float2*)d_in[2];   // frequency_basis     (F,R)   c64
  float* ws = (float*)d_ws;

  lrrt_partial_kernel<<<NB1, 256, 0, stream>>>(rad, att, ws);
  lrrt_reduce_kernel<<<1, 256, 0, stream>>>(ws);
  lrrt_csi_kernel<<<F_SUB / 256, 256, 0, stream>>>(vfb, ws + MOFF, (float*)d_out);
}